// SpikingGCN_26465588478226
// MI455X (gfx1250) — compile-verified
//
#include <hip/hip_runtime.h>

// ---------------------------------------------------------------------------
// SpikingGCN forward for MI455X (gfx1250, wave32, WMMA)
//   N=30000 nodes, E=960000 edges, IN_C=500, HID=128, OUT_C=10, L=2, K=2, T=20
//   Pull-based CSR propagation with precomputed edge weights (no atomics,
//   no dependent scalar chain), WMMA f32 GEMMs, fused LIF.
// ---------------------------------------------------------------------------

#define N_NODES 30000
#define N_EDGES 960000
#define IN_CH   500
#define HID     128
#define OUT_CH  10
#define T_STEPS 20
#define NH      (N_NODES * HID)          // 3,840,000 elements
#define TILES32 ((N_NODES + 31) / 32)    // 938 32-row strips

typedef float v2f __attribute__((ext_vector_type(2)));
typedef float v8f __attribute__((ext_vector_type(8)));

// ---------------------------------------------------------------------------
// CSR build: degree histogram, rsqrt normalization, prefix scan, edge fill
// ---------------------------------------------------------------------------
__global__ void deg_int_kernel(const int* __restrict__ dst, int* __restrict__ ideg) {
    int e = blockIdx.x * blockDim.x + threadIdx.x;
    if (e < N_EDGES) atomicAdd(&ideg[dst[e]], 1);
}

__global__ void dis_kernel(const int* __restrict__ ideg, float* __restrict__ dis) {
    int i = blockIdx.x * blockDim.x + threadIdx.x;
    if (i < N_NODES) dis[i] = rsqrtf((float)ideg[i] + 1.0f);
}

// Single block, 1024 threads: exclusive prefix sum of ideg -> rowptr, fill
__global__ __launch_bounds__(1024)
void scan_kernel(const int* __restrict__ ideg, int* __restrict__ rowptr,
                 int* __restrict__ fill) {
    __shared__ int csum[1024];
    const int t  = threadIdx.x;
    const int CH = (N_NODES + 1023) / 1024;   // 30 elements per thread
    const int beg = t * CH;
    int s = 0;
    for (int j = 0; j < CH; ++j) {
        int idx = beg + j;
        if (idx < N_NODES) s += ideg[idx];
    }
    csum[t] = s;
    __syncthreads();
    for (int off = 1; off < 1024; off <<= 1) {  // Hillis-Steele inclusive scan
        int v = (t >= off) ? csum[t - off] : 0;
        __syncthreads();
        csum[t] += v;
        __syncthreads();
    }
    int run = (t == 0) ? 0 : csum[t - 1];
    for (int j = 0; j < CH; ++j) {
        int idx = beg + j;
        if (idx < N_NODES) {
            rowptr[idx] = run;
            fill[idx]   = run;
            run += ideg[idx];
        }
    }
    if (t == 1023) rowptr[N_NODES] = run;       // == E
}

// Scatter edges into CSR order; precompute full coefficient dis[src]*dis[dst]
__global__ void csr_fill(const int* __restrict__ src, const int* __restrict__ dst,
                         const float* __restrict__ dis,
                         int* __restrict__ fill, int* __restrict__ csr,
                         float* __restrict__ csrw) {
    int e = blockIdx.x * blockDim.x + threadIdx.x;
    if (e >= N_EDGES) return;
    int s = src[e];
    int d = dst[e];
    int pos = atomicAdd(&fill[d], 1);
    csr[pos]  = s;
    csrw[pos] = dis[s] * dis[d];
}

// ---------------------------------------------------------------------------
// Generic fp32 WMMA GEMM (used once for embd = x @ W_in + b_in, K=500)
// ---------------------------------------------------------------------------
__global__ __launch_bounds__(256)
void gemm_wmma_f32(const float* __restrict__ A, const float* __restrict__ W,
                   const float* __restrict__ bias, float* __restrict__ C,
                   int M, int K) {
    const int lane = threadIdx.x & 31;
    const int wave = threadIdx.x >> 5;
    const int lm   = lane & 15;
    const int half = lane >> 4;

    const int tile = blockIdx.x * 8 + wave;
    const int row0 = tile * 16;
    if (row0 >= M) return;                    // whole-wave early out

    v8f acc[8] = {};
    const float* Arow = A + (size_t)(row0 + lm) * (size_t)K + 2 * half;
    for (int k0 = 0; k0 < K; k0 += 4) {
        v2f a = *(const v2f*)(Arow + k0);
        const float* Wk = W + (size_t)(k0 + 2 * half) * HID + lm;
#pragma unroll
        for (int n = 0; n < 8; ++n) {
            v2f b;
            b.x = Wk[n * 16];
            b.y = Wk[HID + n * 16];
            acc[n] = __builtin_amdgcn_wmma_f32_16x16x4_f32(
                false, a, false, b, (short)0, acc[n], false, false);
        }
    }
#pragma unroll
    for (int n = 0; n < 8; ++n) {
        float bcol = bias[n * 16 + lm];
#pragma unroll
        for (int r = 0; r < 8; ++r) {
            C[(size_t)(row0 + r + 8 * half) * HID + n * 16 + lm] = acc[n][r] + bcol;
        }
    }
}

// ---------------------------------------------------------------------------
// Hot fp32 WMMA GEMM, K = HID = 128:  C = A @ W + bias
//   W staged in LDS transposed as float2 pairs with XOR swizzle (64KB exactly,
//   bank-conflict-free ds_load_b64).  One wave: 32x128 strip, B reused by
//   two row strips.
// ---------------------------------------------------------------------------
__global__ __launch_bounds__(256)
void gemm128_wmma(const float* __restrict__ A, const float* __restrict__ W,
                  const float* __restrict__ bias, float* __restrict__ C) {
    __shared__ float2 Wt2[HID * (HID / 2)];   // 8192 pairs = 64KB

    const int tid  = threadIdx.x;
    const int lane = tid & 31;
    const int wave = tid >> 5;
    const int lm   = lane & 15;
    const int half = lane >> 4;

#pragma unroll
    for (int i = 0; i < 32; ++i) {
        int id = i * 256 + tid;               // 8192 pairs total
        int n  = id & 127;
        int k2 = id >> 7;                     // 0..63
        float2 p;
        p.x = W[(size_t)(2 * k2) * HID + n];
        p.y = W[(size_t)(2 * k2 + 1) * HID + n];
        Wt2[n * 64 + (k2 ^ (2 * (n & 15)))] = p;
    }
    __syncthreads();

    const int tile = blockIdx.x * 8 + wave;   // 32-row strip index
    if (tile >= TILES32) return;
    const int r0   = tile * 32;
    const bool two = (r0 + 32) <= N_NODES;

    v8f acc0[8] = {};
    v8f acc1[8] = {};

    const float* Arow0 = A + (size_t)(r0 + lm) * HID + 2 * half;
    const float* Arow1 = A + (size_t)(r0 + 16 + lm) * HID + 2 * half;

    for (int k0 = 0; k0 < HID; k0 += 4) {
        v2f a0 = *(const v2f*)(Arow0 + k0);
        v2f a1 = two ? *(const v2f*)(Arow1 + k0) : a0;
        const int k2 = (k0 >> 1) + half;
        const int swz = 2 * lm;               // n & 15 == lm for every tile
#pragma unroll
        for (int nt = 0; nt < 8; ++nt) {
            float2 bp = Wt2[(nt * 16 + lm) * 64 + (k2 ^ swz)];
            v2f b; b.x = bp.x; b.y = bp.y;
            acc0[nt] = __builtin_amdgcn_wmma_f32_16x16x4_f32(
                false, a0, false, b, (short)0, acc0[nt], false, false);
            if (two)
                acc1[nt] = __builtin_amdgcn_wmma_f32_16x16x4_f32(
                    false, a1, false, b, (short)0, acc1[nt], false, false);
        }
    }

#pragma unroll
    for (int s = 0; s < 2; ++s) {
        if (s == 1 && !two) break;
        const int rbase = r0 + s * 16 + 8 * half;
#pragma unroll
        for (int nt = 0; nt < 8; ++nt) {
            float bcol = bias[nt * 16 + lm];
#pragma unroll
            for (int r = 0; r < 8; ++r) {
                float c = (s == 0 ? acc0[nt][r] : acc1[nt][r]) + bcol;
                C[(size_t)(rbase + r) * HID + nt * 16 + lm] = c;
            }
        }
    }
}

// ---------------------------------------------------------------------------
// Pull-based propagation: one wave per destination node.
//   hout[i,:] = dis[i]^2 * hin[i,:] + sum_e csrw[e] * hin[csr[e],:]
//   Edge index+weight are independent linear scalar loads (no dependent SMEM
//   chain); 2-deep software pipeline issues the next row gather early.
// ---------------------------------------------------------------------------
__global__ __launch_bounds__(256)
void prop_pull(const int* __restrict__ rowptr, const int* __restrict__ csr,
               const float* __restrict__ csrw, const float* __restrict__ dis,
               const float* __restrict__ hin, float* __restrict__ hout) {
    int wid = blockIdx.x * 8 + (threadIdx.x >> 5);
    int i = __builtin_amdgcn_readfirstlane(wid);
    if (i >= N_NODES) return;
    int lane = threadIdx.x & 31;
    const float4* hv = (const float4*)hin;

    float di  = dis[i];
    int   beg = rowptr[i], end = rowptr[i + 1];

    float4 a = hv[(size_t)i * 32 + lane];
    float c0 = di * di;
    float4 acc;
    acc.x = c0 * a.x; acc.y = c0 * a.y; acc.z = c0 * a.z; acc.w = c0 * a.w;

    int e = beg;
    int   sn = (e < end) ? csr[e]  : 0;       // pipelined index
    float wn = (e < end) ? csrw[e] : 0.0f;    // pipelined weight
    float4 vn = hv[(size_t)sn * 32 + lane];   // pipelined row gather
    while (e < end) {
        float  w = wn;
        float4 v = vn;
        ++e;
        sn = (e < end) ? csr[e]  : 0;
        wn = (e < end) ? csrw[e] : 0.0f;
        vn = hv[(size_t)sn * 32 + lane];      // overlap with FMAs below
        acc.x = fmaf(w, v.x, acc.x);
        acc.y = fmaf(w, v.y, acc.y);
        acc.z = fmaf(w, v.z, acc.z);
        acc.w = fmaf(w, v.w, acc.w);
    }
    ((float4*)hout)[(size_t)i * 32 + lane] = acc;
}

// ---------------------------------------------------------------------------
// Hop-2 propagation with fused LIF: result row stays in registers.
//   h = propagate(hin);  v' = 0.5*(v + h);  s = (v' >= 1);  v' = s ? 0 : v'
// ---------------------------------------------------------------------------
__global__ __launch_bounds__(256)
void prop_pull_lif(const int* __restrict__ rowptr, const int* __restrict__ csr,
                   const float* __restrict__ csrw, const float* __restrict__ dis,
                   const float* __restrict__ hin, float* __restrict__ vmem,
                   float* __restrict__ spk, float* __restrict__ sacc) {
    int wid = blockIdx.x * 8 + (threadIdx.x >> 5);
    int i = __builtin_amdgcn_readfirstlane(wid);
    if (i >= N_NODES) return;
    int lane = threadIdx.x & 31;
    const float4* hv = (const float4*)hin;
    const size_t ridx = (size_t)i * 32 + lane;

    float di  = dis[i];
    int   beg = rowptr[i], end = rowptr[i + 1];

    float4 a = hv[ridx];
    float c0 = di * di;
    float4 acc;
    acc.x = c0 * a.x; acc.y = c0 * a.y; acc.z = c0 * a.z; acc.w = c0 * a.w;

    int e = beg;
    int   sn = (e < end) ? csr[e]  : 0;
    float wn = (e < end) ? csrw[e] : 0.0f;
    float4 vn = hv[(size_t)sn * 32 + lane];
    while (e < end) {
        float  w = wn;
        float4 v = vn;
        ++e;
        sn = (e < end) ? csr[e]  : 0;
        wn = (e < end) ? csrw[e] : 0.0f;
        vn = hv[(size_t)sn * 32 + lane];
        acc.x = fmaf(w, v.x, acc.x);
        acc.y = fmaf(w, v.y, acc.y);
        acc.z = fmaf(w, v.z, acc.z);
        acc.w = fmaf(w, v.w, acc.w);
    }

    // Fused LIF
    float4 vo = ((const float4*)vmem)[ridx];
    float4 vnw, sp;
    {
        float vv;
        vv = 0.5f * (vo.x + acc.x); sp.x = (vv >= 1.0f) ? 1.0f : 0.0f; vnw.x = (vv >= 1.0f) ? 0.0f : vv;
        vv = 0.5f * (vo.y + acc.y); sp.y = (vv >= 1.0f) ? 1.0f : 0.0f; vnw.y = (vv >= 1.0f) ? 0.0f : vv;
        vv = 0.5f * (vo.z + acc.z); sp.z = (vv >= 1.0f) ? 1.0f : 0.0f; vnw.z = (vv >= 1.0f) ? 0.0f : vv;
        vv = 0.5f * (vo.w + acc.w); sp.w = (vv >= 1.0f) ? 1.0f : 0.0f; vnw.w = (vv >= 1.0f) ? 0.0f : vv;
    }
    ((float4*)vmem)[ridx] = vnw;
    ((float4*)spk)[ridx]  = sp;
    if (sacc) {
        float4 ac = ((const float4*)sacc)[ridx];
        ac.x += sp.x; ac.y += sp.y; ac.z += sp.z; ac.w += sp.w;
        ((float4*)sacc)[ridx] = ac;
    }
}

// ---------------------------------------------------------------------------
// Output head: out[i,o] = (sacc[i,:] . W_out[:,o]) / T + b_out[o]
// ---------------------------------------------------------------------------
__global__ void out_kernel(const float* __restrict__ sacc, const float* __restrict__ Wout,
                           const float* __restrict__ bout, float* __restrict__ out) {
    int i = blockIdx.x * blockDim.x + threadIdx.x;
    if (i >= N_NODES * OUT_CH) return;
    int node = i / OUT_CH;
    int o    = i % OUT_CH;
    const float* r = sacc + (size_t)node * HID;
    float sum = 0.0f;
#pragma unroll 8
    for (int k = 0; k < HID; ++k) sum += r[k] * Wout[k * OUT_CH + o];
    out[i] = sum * (1.0f / T_STEPS) + bout[o];
}

// ---------------------------------------------------------------------------
// Host-side orchestration (graph-capture safe: async ops on `stream` only)
// ---------------------------------------------------------------------------
extern "C" void kernel_launch(void* const* d_in, const int* in_sizes, int n_in,
                              void* d_out, int out_size, void* d_ws, size_t ws_size,
                              hipStream_t stream) {
    const float* x     = (const float*)d_in[0];   // [N, 500]
    const float* W_in  = (const float*)d_in[1];   // [500, 128]
    const float* b_in  = (const float*)d_in[2];   // [128]
    const float* Wc    = (const float*)d_in[3];   // [2, 128, 128]
    const float* bc    = (const float*)d_in[4];   // [2, 128]
    const float* W_out = (const float*)d_in[5];   // [128, 10]
    const float* b_out = (const float*)d_in[6];   // [10]
    const int*   eidx  = (const int*)d_in[7];     // [2, E] int32
    const int*   src   = eidx;
    const int*   dst   = eidx + N_EDGES;

    // Workspace layout (4-byte words)
    char* wsb = (char*)d_ws;
    size_t off = 0;
    auto alloc = [&](size_t words) { void* p = wsb + off; off += ((words + 15) & ~(size_t)15) * 4; return p; };
    int*   ideg   = (int*)  alloc(N_NODES);
    int*   rowptr = (int*)  alloc(N_NODES + 1);
    int*   fill   = (int*)  alloc(N_NODES);
    int*   csr    = (int*)  alloc(N_EDGES);
    float* csrw   = (float*)alloc(N_EDGES);
    float* dis    = (float*)alloc(N_NODES);
    float* embd   = (float*)alloc(NH);
    float* bufA   = (float*)alloc(NH);
    float* bufB   = (float*)alloc(NH);
    float* v0     = (float*)alloc(NH);
    float* v1     = (float*)alloc(NH);   // contiguous with v0
    float* spk    = (float*)alloc(NH);
    float* sacc   = (float*)alloc(NH);

    // Zero state every call (deterministic across graph replays)
    hipMemsetAsync(ideg, 0, (size_t)N_NODES * sizeof(int), stream);
    hipMemsetAsync(v0,   0, (size_t)2 * NH * sizeof(float), stream);
    hipMemsetAsync(sacc, 0, (size_t)NH * sizeof(float), stream);

    // CSR build + normalization (weights folded into CSR entries)
    deg_int_kernel<<<(N_EDGES + 255) / 256, 256, 0, stream>>>(dst, ideg);
    dis_kernel<<<(N_NODES + 255) / 256, 256, 0, stream>>>(ideg, dis);
    scan_kernel<<<1, 1024, 0, stream>>>(ideg, rowptr, fill);
    csr_fill<<<(N_EDGES + 255) / 256, 256, 0, stream>>>(src, dst, dis, fill, csr, csrw);

    // Loop-invariant input embedding: embd = x @ W_in + b_in  (K=500)
    gemm_wmma_f32<<<(N_NODES / 16 + 7) / 8, 256, 0, stream>>>(
        x, W_in, b_in, embd, N_NODES, IN_CH);

    const int g128_blocks = (TILES32 + 7) / 8;          // 118
    const int pull_blocks = (N_NODES + 7) / 8;          // 3750 (one wave/node)

    for (int t = 0; t < T_STEPS; ++t) {
        for (int l = 0; l < 2; ++l) {
            const float* hin = (l == 0) ? embd : spk;
            // bufA = hin @ Wc[l] + bc[l]
            gemm128_wmma<<<g128_blocks, 256, 0, stream>>>(
                hin, Wc + (size_t)l * HID * HID, bc + (size_t)l * HID, bufA);
            // hop 1 (pull): bufB = P(bufA)
            prop_pull<<<pull_blocks, 256, 0, stream>>>(
                rowptr, csr, csrw, dis, bufA, bufB);
            // hop 2 (pull) + fused LIF: v_l, spk, (sacc for l==1)
            prop_pull_lif<<<pull_blocks, 256, 0, stream>>>(
                rowptr, csr, csrw, dis, bufB, (l == 0) ? v0 : v1, spk,
                (l == 1) ? sacc : (float*)nullptr);
        }
    }

    // out = (sacc/T) @ W_out + b_out
    out_kernel<<<(N_NODES * OUT_CH + 255) / 256, 256, 0, stream>>>(
        sacc, W_out, b_out, (float*)d_out);
}